// CrossLayer_6322191860059
// MI455X (gfx1250) — compile-verified
//
#include <hip/hip_runtime.h>

// CrossLayer (BatchNorm1d + rank-1 cross) for MI455X (gfx1250, wave32).
//
// B=16384, D=1024, fp32 (matches reference). Bandwidth-bound:
// ~0.1 GFLOP vs ~192MB min traffic -> ~11us @ 23.3 TB/s HBM; working set
// (x + x0 + out = 192MB) fits the 192MB global L2, so the second pass over x
// should mostly hit L2.
//
// Pipeline:
//   k0: zero column-stat accumulators in workspace
//   k1: one pass over x -> per-column sum / sum-of-squares (global f32 atomics)
//   k2: finalize scale[d]=gamma*rsqrt(var+eps), shift[d]=beta-mean*scale,
//       wscale[d]=weight*scale, C = sum(shift*weight)   (1 block)
//   k3: fused GEMV (V_WMMA_F32_16X16X4_F32) + elementwise output pass.
//       B tile holds wscale replicated across ALL 16 columns, so every
//       column of D accumulates the same row-dot => branchless, EXEC-uniform
//       inner loop (no per-iteration lane masking).

typedef __attribute__((ext_vector_type(2))) float v2f;
typedef __attribute__((ext_vector_type(4))) float v4f;
typedef __attribute__((ext_vector_type(8))) float v8f;

#define EPSF 1e-8f

// workspace layout in floats (total 5121 floats ~= 20.5 KB)
#define WS_SUMS   0
#define WS_SUMSQ  1024
#define WS_SCALE  2048
#define WS_SHIFT  3072
#define WS_WSCALE 4096
#define WS_CCONST 5120

__global__ __launch_bounds__(256)
void k0_zero(float* __restrict__ ws, int n) {
    for (int i = blockIdx.x * blockDim.x + threadIdx.x; i < n;
         i += gridDim.x * blockDim.x)
        ws[i] = 0.0f;
}

// One block = 64 rows; 256 threads * float4 = one full row of D=1024 columns.
__global__ __launch_bounds__(256)
void k1_colstats(const float* __restrict__ x, float* __restrict__ ws,
                 int B, int D) {
    const int col = threadIdx.x * 4;          // D == 1024 == 256*4
    const int r0  = blockIdx.x * 64;
    float s1[4] = {0.f, 0.f, 0.f, 0.f};
    float s2[4] = {0.f, 0.f, 0.f, 0.f};
    for (int r = 0; r < 64; ++r) {
        v4f v = *(const v4f*)(x + (size_t)(r0 + r) * D + col);
        #pragma unroll
        for (int j = 0; j < 4; ++j) {
            s1[j] += v[j];
            s2[j] += v[j] * v[j];
        }
    }
    #pragma unroll
    for (int j = 0; j < 4; ++j) {
        atomicAdd(&ws[WS_SUMS  + col + j], s1[j]);
        atomicAdd(&ws[WS_SUMSQ + col + j], s2[j]);
    }
}

// Single block, D threads (1024 = 32 waves).
__global__ __launch_bounds__(1024)
void k2_finalize(const float* __restrict__ gamma, const float* __restrict__ beta,
                 const float* __restrict__ weight, float* __restrict__ ws,
                 int B) {
    __shared__ float red[1024];
    const int d = threadIdx.x;
    const float invB = 1.0f / (float)B;
    float mean = ws[WS_SUMS + d] * invB;
    float var  = ws[WS_SUMSQ + d] * invB - mean * mean;
    float sc   = gamma[d] * rsqrtf(var + EPSF);
    float sh   = beta[d] - mean * sc;
    float w    = weight[d];
    ws[WS_SCALE  + d] = sc;
    ws[WS_SHIFT  + d] = sh;
    ws[WS_WSCALE + d] = w * sc;
    red[d] = sh * w;
    __syncthreads();
    for (int off = 512; off > 0; off >>= 1) {
        if (d < off) red[d] += red[d + off];
        __syncthreads();
    }
    if (d == 0) ws[WS_CCONST] = red[0];
}

// 256 threads = 8 waves; wave handles 16 rows -> 128 rows/block.
__global__ __launch_bounds__(256)
void k3_main(const float* __restrict__ x, const float* __restrict__ x0,
             const float* __restrict__ bias, const float* __restrict__ ws,
             float* __restrict__ out, int D) {
    const int tid  = threadIdx.x;
    const int lane = tid & 31;
    const int wave = tid >> 5;
    const int blockRow0 = blockIdx.x * 128;
    const int r0 = blockRow0 + wave * 16;

    // ---- GEMV: s[r0..r0+15] = x_rows . wscale via V_WMMA_F32_16X16X4_F32 ----
    // A layout (32-bit, 16x4 MxK): lanes 0-15  -> M=lane,    VGPR0=K0, VGPR1=K1
    //                              lanes 16-31 -> M=lane-16, VGPR0=K2, VGPR1=K3
    // B layout (4x16 KxN): VGPR0: lanes0-15 = row K0, lanes16-31 = row K1
    //                      VGPR1: lanes0-15 = row K2, lanes16-31 = row K3
    // We replicate wscale[kb+k] across all N, so lane L needs:
    //   b.x = wsc[kb + hl], b.y = wsc[kb + 2 + hl]   with hl = L>>4.
    // Every column of D then accumulates the identical dot product.
    const int hl = lane >> 4;
    const float* __restrict__ xrow =
        x + (size_t)(r0 + (lane & 15)) * D + (hl << 1);
    const float* __restrict__ wp = ws + WS_WSCALE + hl;

    v8f c = {0.f, 0.f, 0.f, 0.f, 0.f, 0.f, 0.f, 0.f};
    #pragma unroll 4
    for (int kb = 0; kb < D; kb += 4) {
        v2f a = *(const v2f*)(xrow + kb);   // x[row][kb+2*hl .. +1]
        v2f b;
        b.x = wp[kb];                       // wscale[kb + hl]
        b.y = wp[kb + 2];                   // wscale[kb + 2 + hl]
        // 8 args: (neg_a, A, neg_b, B, c_mod, C, reuse_a, reuse_b)
        c = __builtin_amdgcn_wmma_f32_16x16x4_f32(
                false, a, false, b, (short)0, c, false, false);
    }

    // All columns equal; take N=0: lane0 VGPR i = s[r0+i] (M=0..7),
    //                              lane16 VGPR i = s[r0+8+i] (M=8..15).
    __shared__ float s_lds[128];
    const float Cc = ws[WS_CCONST];
    if (lane == 0) {
        #pragma unroll
        for (int i = 0; i < 8; ++i) s_lds[wave * 16 + i] = c[i] + Cc;
    }
    if (lane == 16) {
        #pragma unroll
        for (int i = 0; i < 8; ++i) s_lds[wave * 16 + 8 + i] = c[i] + Cc;
    }
    __syncthreads();

    // ---- fused output pass: out = x0*s + bias + (x*scale + shift) ----
    const int col = tid * 4;                  // D == 1024 == 256*4
    v4f sc = *(const v4f*)(ws + WS_SCALE + col);
    v4f sh = *(const v4f*)(ws + WS_SHIFT + col);
    v4f bi = *(const v4f*)(bias + col);
    for (int r = 0; r < 128; ++r) {
        const size_t off = (size_t)(blockRow0 + r) * D + col;
        v4f xv  = *(const v4f*)(x  + off);
        v4f x0v = *(const v4f*)(x0 + off);
        const float s = s_lds[r];
        v4f o;
        #pragma unroll
        for (int j = 0; j < 4; ++j)
            o[j] = x0v[j] * s + bi[j] + (xv[j] * sc[j] + sh[j]);
        *(v4f*)(out + off) = o;
    }
}

extern "C" void kernel_launch(void* const* d_in, const int* in_sizes, int n_in,
                              void* d_out, int out_size, void* d_ws, size_t ws_size,
                              hipStream_t stream) {
    const float* x      = (const float*)d_in[0];
    const float* x0     = (const float*)d_in[1];
    const float* gamma  = (const float*)d_in[2];
    const float* beta   = (const float*)d_in[3];
    const float* weight = (const float*)d_in[4];
    const float* bias   = (const float*)d_in[5];
    float* out = (float*)d_out;
    float* ws  = (float*)d_ws;

    const int D = in_sizes[2];          // 1024
    const int B = in_sizes[0] / D;      // 16384

    k0_zero    <<<8,       256, 0, stream>>>(ws, 2 * D);
    k1_colstats<<<B / 64,  256, 0, stream>>>(x, ws, B, D);
    k2_finalize<<<1,       D,   0, stream>>>(gamma, beta, weight, ws, B);
    k3_main    <<<B / 128, 256, 0, stream>>>(x, x0, bias, ws, out, D);
}